// LogNeuralBackflow_62268435857712
// MI455X (gfx1250) — compile-verified
//
#include <hip/hip_runtime.h>
#include <math.h>

#define BF_SIZE  256   // spin-orbitals
#define BF_NF    128   // fermions
#define BF_HID   512   // hidden units
#define BF_BATCH 512
#define BF_NOUT  (BF_SIZE * BF_NF)   // 32768

typedef __attribute__((ext_vector_type(2))) float v2f;
typedef __attribute__((ext_vector_type(8))) float v8f;

// ---------------------------------------------------------------------------
// Kernel 1: occupied-index bookkeeping + hidden layer H = tanh(n@W0 + b0).
// n@W0 is a gather-sum of the 128 occupied rows of W0 (n is 0/1).
// Writes planar Hr/Hi (f32) and invR (r -> position in R, or -1) to workspace.
// ---------------------------------------------------------------------------
__global__ void nbf_backflow_hidden(const int* __restrict__ n,
                                    const float* __restrict__ W0,   // (SIZE,HID) cplx interleaved
                                    const float* __restrict__ b0,   // (HID,) cplx interleaved
                                    float* __restrict__ Hr,
                                    float* __restrict__ Hi,
                                    int* __restrict__ invR) {
  const int b = blockIdx.x;
  __shared__ int Rsh[BF_NF];
  __shared__ int invSh[BF_SIZE];
  if (threadIdx.x == 0) {
    int cnt = 0;
    for (int s = 0; s < BF_SIZE; ++s) {
      if (n[b * BF_SIZE + s] > 0) { Rsh[cnt] = s; invSh[s] = cnt; ++cnt; }
      else invSh[s] = -1;
    }
  }
  __syncthreads();
  for (int s = threadIdx.x; s < BF_SIZE; s += blockDim.x)
    invR[b * BF_SIZE + s] = invSh[s];

  for (int h = threadIdx.x; h < BF_HID; h += blockDim.x) {
    float ar = b0[2 * h], ai = b0[2 * h + 1];
    for (int t = 0; t < BF_NF; ++t) {
      const int s = Rsh[t];
      const float* w = W0 + 2 * ((size_t)s * BF_HID + h);
      ar += w[0];
      ai += w[1];
    }
    // complex tanh: tanh(x+iy) = (sinh 2x + i sin 2y) / (cosh 2x + cos 2y)
    const float x2 = 2.0f * ar, y2 = 2.0f * ai;
    const float d = coshf(x2) + cosf(y2);
    const float id = 1.0f / d;
    Hr[(size_t)b * BF_HID + h] = sinhf(x2) * id;
    Hi[(size_t)b * BF_HID + h] = sinf(y2) * id;
  }
}

// ---------------------------------------------------------------------------
// Kernel 2: complex GEMM F = H @ Wout using V_WMMA_F32_16X16X4_F32, fused with
// the gather/epilogue: A[b, invR[b][r], j] = M[r,j] + bout[c] + F[b,c],
// c = r*NF + j.  Register-blocked 2x in M: each wave32 computes a
// 32(batch) x 16(column) tile, so every B (Wout) load feeds 8 WMMAs.
// Dr = acc(Ar*Br) - acc(Ai*Bi);  Di = acc(Ar*Bi) + acc(Ai*Br).
// ---------------------------------------------------------------------------
__global__ void nbf_gemm_wmma(const float* __restrict__ Hr,
                              const float* __restrict__ Hi,
                              const float* __restrict__ Wout,  // (HID,NOUT) cplx interleaved
                              const float* __restrict__ bout,  // (NOUT,)   cplx interleaved
                              const float* __restrict__ M,     // (SIZE,NF) cplx interleaved
                              const int* __restrict__ invR,
                              float* __restrict__ Aout) {      // (B,NF,NF) cplx interleaved
  const int lane   = threadIdx.x & 31;
  const int wave   = threadIdx.x >> 5;
  const int lane16 = lane & 15;
  const int half   = lane >> 4;                 // 0: lanes 0-15, 1: lanes 16-31
  const int n0 = (blockIdx.x * 8 + wave) * 16;  // output-column tile base
  const int m0 = blockIdx.y * 32;               // batch tile base (2 x 16)

  v8f accRR0 = {}, accII0 = {}, accI0 = {};     // tile 0: batches m0 .. m0+15
  v8f accRR1 = {}, accII1 = {}, accI1 = {};     // tile 1: batches m0+16 .. m0+31

  // A-matrix (16x4 f32): lanes 0-15 carry K=k,k+1 of row m0+lane;
  //                      lanes 16-31 carry K=k+2,k+3.
  const float* hrp0 = Hr + (size_t)(m0 + lane16) * BF_HID;
  const float* hip0 = Hi + (size_t)(m0 + lane16) * BF_HID;
  const float* hrp1 = Hr + (size_t)(m0 + 16 + lane16) * BF_HID;
  const float* hip1 = Hi + (size_t)(m0 + 16 + lane16) * BF_HID;
  const int nCol = n0 + lane16;

  for (int kk = 0; kk < BF_HID; kk += 4) {
    const int kc = kk + 2 * half;
    const v2f ar0 = *(const v2f*)(hrp0 + kc);
    const v2f ai0 = *(const v2f*)(hip0 + kc);
    const v2f ar1 = *(const v2f*)(hrp1 + kc);
    const v2f ai1 = *(const v2f*)(hip1 + kc);

    // B-matrix (4x16 f32): VGPR0 holds K=kc, VGPR1 holds K=kc+1, N=lane16.
    const float* w0p = Wout + 2 * ((size_t)kc * BF_NOUT + nCol);
    const float* w1p = Wout + 2 * ((size_t)(kc + 1) * BF_NOUT + nCol);
    const v2f c0 = *(const v2f*)w0p;  // (re, im) of Wout[kc][n]
    const v2f c1 = *(const v2f*)w1p;  // (re, im) of Wout[kc+1][n]
    const v2f br = {c0.x, c1.x};
    const v2f bi = {c0.y, c1.y};

    // prefetch Wout rows four K-steps ahead
    __builtin_prefetch((const void*)(w0p + 8 * (size_t)BF_NOUT), 0, 1);

    accRR0 = __builtin_amdgcn_wmma_f32_16x16x4_f32(false, ar0, false, br,
                                                   (short)0, accRR0, false, false);
    accII0 = __builtin_amdgcn_wmma_f32_16x16x4_f32(false, ai0, false, bi,
                                                   (short)0, accII0, false, false);
    accI0  = __builtin_amdgcn_wmma_f32_16x16x4_f32(false, ar0, false, bi,
                                                   (short)0, accI0,  false, false);
    accI0  = __builtin_amdgcn_wmma_f32_16x16x4_f32(false, ai0, false, br,
                                                   (short)0, accI0,  false, false);

    accRR1 = __builtin_amdgcn_wmma_f32_16x16x4_f32(false, ar1, false, br,
                                                   (short)0, accRR1, false, false);
    accII1 = __builtin_amdgcn_wmma_f32_16x16x4_f32(false, ai1, false, bi,
                                                   (short)0, accII1, false, false);
    accI1  = __builtin_amdgcn_wmma_f32_16x16x4_f32(false, ar1, false, bi,
                                                   (short)0, accI1,  false, false);
    accI1  = __builtin_amdgcn_wmma_f32_16x16x4_f32(false, ai1, false, br,
                                                   (short)0, accI1,  false, false);
  }

  // Epilogue: whole tile maps to one orbital row r (NF is a multiple of 16).
  const int r = n0 / BF_NF;
  const int j = (n0 % BF_NF) + lane16;
  const v2f mv = *(const v2f*)(M + 2 * ((size_t)r * BF_NF + j));
  const v2f bo = *(const v2f*)(bout + 2 * (size_t)(n0 + lane16));
  const float baseRe = mv.x + bo.x;
  const float baseIm = mv.y + bo.y;

  // C/D layout: VGPR g -> row m = g + 8*half, column = lane16.
  for (int g = 0; g < 8; ++g) {
    {
      const int b = m0 + g + 8 * half;
      const int iv = invR[b * BF_SIZE + r];
      if (iv >= 0) {
        v2f val;
        val.x = baseRe + accRR0[g] - accII0[g];
        val.y = baseIm + accI0[g];
        v2f* dst = (v2f*)(Aout + 2 * ((size_t)b * (BF_NF * BF_NF) +
                                      (size_t)iv * BF_NF + j));
        *dst = val;
      }
    }
    {
      const int b = m0 + 16 + g + 8 * half;
      const int iv = invR[b * BF_SIZE + r];
      if (iv >= 0) {
        v2f val;
        val.x = baseRe + accRR1[g] - accII1[g];
        val.y = baseIm + accI1[g];
        v2f* dst = (v2f*)(Aout + 2 * ((size_t)b * (BF_NF * BF_NF) +
                                      (size_t)iv * BF_NF + j));
        *dst = val;
      }
    }
  }
}

// ---------------------------------------------------------------------------
// Kernel 3: per-batch complex LU with partial pivoting in LDS (128x129 padded
// rows -> ~132KB dynamic LDS, fits CDNA5's 320KB WGP LDS), then
// out[b] = log|det| + i*arg(det) with Im wrapped to (-pi, pi].
// ---------------------------------------------------------------------------
__global__ void nbf_lu_logdet(const float* __restrict__ Aout,
                              float* __restrict__ out) {
  const int LDA = BF_NF + 1;  // pad to dodge LDS bank conflicts on column walks
  extern __shared__ unsigned char smem[];
  float2* A   = (float2*)smem;                 // 128 x 129 complex
  float* piv  = (float*)(A + BF_NF * LDA);     // 128 pivot magnitudes
  int*   meta = (int*)(piv + BF_NF);           // [0]=pivot row, [1]=swap count

  const int b = blockIdx.x;
  const int tid = threadIdx.x;                 // 128 threads; thread i owns row i

  const float2* src = (const float2*)Aout + (size_t)b * BF_NF * BF_NF;
  for (int idx = tid; idx < BF_NF * BF_NF; idx += blockDim.x)
    A[(idx / BF_NF) * LDA + (idx % BF_NF)] = src[idx];
  if (tid == 0) meta[1] = 0;
  __syncthreads();

  for (int k = 0; k < BF_NF; ++k) {
    const float2 vk = A[tid * LDA + k];
    piv[tid] = (tid >= k) ? (vk.x * vk.x + vk.y * vk.y) : -1.0f;
    __syncthreads();
    if (tid == 0) {
      int p = k;
      float best = piv[k];
      for (int i = k + 1; i < BF_NF; ++i)
        if (piv[i] > best) { best = piv[i]; p = i; }
      meta[0] = p;
      if (p != k) meta[1]++;
    }
    __syncthreads();
    const int p = meta[0];
    if (p != k) {  // swap rows k <-> p, threads across columns
      const float2 t = A[k * LDA + tid];
      A[k * LDA + tid] = A[p * LDA + tid];
      A[p * LDA + tid] = t;
    }
    __syncthreads();
    const float2 akk = A[k * LDA + k];
    const float den = akk.x * akk.x + akk.y * akk.y;
    const float2 inv = {akk.x / den, -akk.y / den};
    if (tid > k) {
      const float2 aik = A[tid * LDA + k];
      const float2 l = {aik.x * inv.x - aik.y * inv.y,
                        aik.x * inv.y + aik.y * inv.x};
      A[tid * LDA + k] = l;
      for (int j = k + 1; j < BF_NF; ++j) {
        const float2 u = A[k * LDA + j];  // broadcast read
        float2 a = A[tid * LDA + j];
        a.x -= l.x * u.x - l.y * u.y;
        a.y -= l.x * u.y + l.y * u.x;
        A[tid * LDA + j] = a;
      }
    }
    __syncthreads();
  }

  if (tid == 0) {
    float sre = 0.0f, sim = 0.0f;
    for (int k = 0; k < BF_NF; ++k) {
      const float2 u = A[k * LDA + k];
      sre += 0.5f * logf(u.x * u.x + u.y * u.y);
      sim += atan2f(u.y, u.x);
    }
    const float PI_F = 3.14159265358979f, TWO_PI_F = 6.28318530717959f;
    sim += (float)meta[1] * PI_F;     // each row swap multiplies det by -1
    sim = fmodf(sim, TWO_PI_F);
    if (sim > PI_F) sim -= TWO_PI_F;
    else if (sim <= -PI_F) sim += TWO_PI_F;
    out[2 * b]     = sre;             // log|det|
    out[2 * b + 1] = sim;             // arg(det) in (-pi, pi]
  }
}

// ---------------------------------------------------------------------------
extern "C" void kernel_launch(void* const* d_in, const int* in_sizes, int n_in,
                              void* d_out, int out_size, void* d_ws, size_t ws_size,
                              hipStream_t stream) {
  const int*   n    = (const int*)d_in[0];
  const float* M    = (const float*)d_in[1];   // complex64 interleaved
  const float* W0   = (const float*)d_in[2];
  const float* b0   = (const float*)d_in[3];
  const float* Wout = (const float*)d_in[4];
  const float* bout = (const float*)d_in[5];
  float* out = (float*)d_out;

  // Workspace layout (floats): Hr | Hi | invR (as ints) | A (gathered matrices)
  float* wsf = (float*)d_ws;
  float* Hr = wsf;
  float* Hi = wsf + (size_t)BF_BATCH * BF_HID;
  int* invR = (int*)(wsf + 2 * (size_t)BF_BATCH * BF_HID);
  float* Aout = wsf + 2 * (size_t)BF_BATCH * BF_HID + (size_t)BF_BATCH * BF_SIZE;

  hipLaunchKernelGGL(nbf_backflow_hidden, dim3(BF_BATCH), dim3(256), 0, stream,
                     n, W0, b0, Hr, Hi, invR);

  // 32x16 tile per wave, 8 waves/block: grid = (NOUT/128, BATCH/32)
  hipLaunchKernelGGL(nbf_gemm_wmma, dim3(BF_NOUT / 128, BF_BATCH / 32),
                     dim3(256), 0, stream, Hr, Hi, Wout, bout, M, invR, Aout);

  const size_t sh = (size_t)BF_NF * (BF_NF + 1) * sizeof(float2) +
                    BF_NF * sizeof(float) + 2 * sizeof(int);
  hipLaunchKernelGGL(nbf_lu_logdet, dim3(BF_BATCH), dim3(BF_NF), sh, stream,
                     Aout, out);
}